// ParaInfNeuron_Text_55645596287070
// MI455X (gfx1250) — compile-verified
//
#include <hip/hip_runtime.h>

// ParaInfNeuron elementwise kernel for MI455X (gfx1250).
//
// out[i] = 0.25 * sum_{t=1..4} ( x[i]*(4/t) + 0.5/t >= 1.0 ? 1.0 : 0.0 )
//
// Pure streaming op: 128 MiB in + 128 MiB out, no reuse -> HBM-bound
// (~11.5 us at 23.3 TB/s). Strategy: b128 vector loads/stores with
// non-temporal hints (stream does not fit the 192 MB L2 and is touched
// once), 256-thread blocks (8 wave32), 4 vectors per thread for MLP.

#define BLOCK_THREADS 256
#define VEC_PER_THREAD 4

// Native clang vector (NOT HIP_vector_type) so nontemporal builtins accept it.
typedef __attribute__((ext_vector_type(4))) float v4f;

// Reproduce the reference's arithmetic exactly: coef = 4 * fl(1/t),
// bias = 0.5 * fl(1/t), compare (x * coef) + bias >= 1.0 with SEPARATE
// mul and add roundings. contract(off) stops hipcc's default
// -ffp-contract=fast from fusing the t=3 term (c = fl(4/3)) into an FMA,
// which could flip the >=1.0 spike decision within 1 ulp of the threshold.
__device__ __forceinline__ float neuron_elem(float x) {
#pragma clang fp contract(off)
    // fl(1/t) for t = 1..4 (1/3 rounds to 0x3EAAAAAB; others exact)
    const float inv1 = 1.0f;
    const float inv2 = 0.5f;
    const float inv3 = 0.33333334f; // fl(1/3)
    const float inv4 = 0.25f;

    const float c1 = 4.0f * inv1, b1 = 0.5f * inv1;
    const float c2 = 4.0f * inv2, b2 = 0.5f * inv2;
    const float c3 = 4.0f * inv3, b3 = 0.5f * inv3;
    const float c4 = 4.0f * inv4, b4 = 0.5f * inv4;

    float acc = 0.0f;
    acc += ((x * c1) + b1 >= 1.0f) ? 1.0f : 0.0f;
    acc += ((x * c2) + b2 >= 1.0f) ? 1.0f : 0.0f;
    acc += ((x * c3) + b3 >= 1.0f) ? 1.0f : 0.0f;
    acc += ((x * c4) + b4 >= 1.0f) ? 1.0f : 0.0f;
    return acc * 0.25f;
}

__global__ __launch_bounds__(BLOCK_THREADS)
void parainf_neuron_kernel(const v4f* __restrict__ x,
                           v4f* __restrict__ out,
                           int n4) {
    const int base = blockIdx.x * (BLOCK_THREADS * VEC_PER_THREAD) + threadIdx.x;

#pragma unroll
    for (int k = 0; k < VEC_PER_THREAD; ++k) {
        const int i = base + k * BLOCK_THREADS;
        if (i < n4) {
            // NT load: single-touch stream, deprioritize cache residency.
            v4f v = __builtin_nontemporal_load(x + i);
            v4f r;
            r.x = neuron_elem(v.x);
            r.y = neuron_elem(v.y);
            r.z = neuron_elem(v.z);
            r.w = neuron_elem(v.w);
            __builtin_nontemporal_store(r, out + i);
        }
    }
}

// Tail kernel for element counts not divisible by 4 (not hit with the
// reference shape 4*2048*4096, but keeps the launch generic/deterministic).
__global__ __launch_bounds__(BLOCK_THREADS)
void parainf_neuron_tail_kernel(const float* __restrict__ x,
                                float* __restrict__ out,
                                int start, int n) {
    const int i = start + blockIdx.x * BLOCK_THREADS + threadIdx.x;
    if (i < n) {
        out[i] = neuron_elem(x[i]);
    }
}

extern "C" void kernel_launch(void* const* d_in, const int* in_sizes, int n_in,
                              void* d_out, int out_size, void* d_ws, size_t ws_size,
                              hipStream_t stream) {
    (void)n_in; (void)d_ws; (void)ws_size; (void)out_size;

    const float* x = (const float*)d_in[0];
    float* out = (float*)d_out;
    const int n = in_sizes[0];          // 4*2048*4096 = 33,554,432
    const int n4 = n >> 2;              // float4-vector count

    const int per_block = BLOCK_THREADS * VEC_PER_THREAD;
    const int grid = (n4 + per_block - 1) / per_block;
    if (grid > 0) {
        parainf_neuron_kernel<<<grid, BLOCK_THREADS, 0, stream>>>(
            (const v4f*)x, (v4f*)out, n4);
    }

    const int tail_start = n4 << 2;
    const int tail = n - tail_start;
    if (tail > 0) {
        const int tgrid = (tail + BLOCK_THREADS - 1) / BLOCK_THREADS;
        parainf_neuron_tail_kernel<<<tgrid, BLOCK_THREADS, 0, stream>>>(
            x, out, tail_start, n);
    }
}